// GCN_33054068310209
// MI455X (gfx1250) — compile-verified
//
#include <hip/hip_runtime.h>
#include <hip/hip_bf16.h>

typedef __attribute__((ext_vector_type(2))) float v2f;
typedef __attribute__((ext_vector_type(8))) float v8f;

constexpr int N_NODES = 100000;
constexpr int N_EDGES = 1600000;
constexpr int FT      = 256;
constexpr int HID     = 128;
constexpr int NCLS    = 10;
constexpr int NPAD    = 16;     // NCLS padded to WMMA tile width
constexpr int N_TRAIN = 5000;

// ---------------------------------------------------------------------------
// Setup: pad W2 [128x10] -> W2p [128x16] (zero-filled), zero the scalar output.
// ---------------------------------------------------------------------------
__global__ void gcn_setup(const float* __restrict__ W2, float* __restrict__ W2p,
                          float* __restrict__ out) {
  int t = blockIdx.x * blockDim.x + threadIdx.x;
  if (t == 0) out[0] = 0.0f;
  if (t < HID * NPAD) {
    int k = t / NPAD, n = t % NPAD;
    W2p[t] = (n < NCLS) ? W2[k * NCLS + n] : 0.0f;
  }
}

// ---------------------------------------------------------------------------
// Init an [N_NODES x width] accumulator to a broadcast bias (pad cols -> 0).
// Folds the "+ b" of the reference into the segment-sum accumulator.
// ---------------------------------------------------------------------------
__global__ void gcn_init_bias(float* __restrict__ S, const float* __restrict__ bias,
                              int width, int nvalid, long long total) {
  long long t = (long long)blockIdx.x * blockDim.x + threadIdx.x;
  if (t >= total) return;
  int j = (int)(t % width);
  S[t] = (j < nvalid) ? bias[j] : 0.0f;
}

// ---------------------------------------------------------------------------
// GEMM1: H[100000x128] = X[100000x256] @ W1[256x128], fp32 WMMA 16x16x4.
// One wave = one 16x16 tile. Block = 256 threads = 8 waves = 8 N-tiles
// (HID/16 == 8), grid.x = M tiles (100000/16 = 6250, exact).
// A layout (ISA 7.12.2, 32-bit A 16x4): lanes 0-15 hold K=k0,k0+1; lanes
// 16-31 hold K=k0+2,k0+3 for row M=lane%16. B/C/D row-striped across lanes.
// ---------------------------------------------------------------------------
__global__ void gcn_gemm1_wmma(const float* __restrict__ X,
                               const float* __restrict__ W1,
                               float* __restrict__ H) {
  const int lane = threadIdx.x & 31;
  const int wv   = threadIdx.x >> 5;   // wave32: 8 waves per block
  const int half = lane >> 4;          // 0: lanes 0-15, 1: lanes 16-31
  const int l    = lane & 15;
  const int m0   = blockIdx.x * 16;
  const int n0   = wv * 16;

  const float* xrow = X + (long long)(m0 + l) * FT;
  v8f c = {0.f, 0.f, 0.f, 0.f, 0.f, 0.f, 0.f, 0.f};

  for (int k0 = 0; k0 < FT; k0 += 4) {
    const int ka = k0 + 2 * half;
    v2f a;
    a.x = xrow[ka];
    a.y = xrow[ka + 1];
    v2f b;
    b.x = W1[(long long)ka * HID + n0 + l];
    b.y = W1[(long long)(ka + 1) * HID + n0 + l];
    c = __builtin_amdgcn_wmma_f32_16x16x4_f32(false, a, false, b,
                                              (short)0, c, false, false);
  }
#pragma unroll
  for (int r = 0; r < 8; ++r)
    H[(long long)(m0 + half * 8 + r) * HID + n0 + l] = c[r];
}

// ---------------------------------------------------------------------------
// SpMM layer 1 (width 128): S[row] += val * H[col].  32 lanes per edge,
// 4 columns per lane. H and S live in L2 (51 MB each << 192 MB).
// ---------------------------------------------------------------------------
__global__ void gcn_spmm_hid(const int* __restrict__ rows,
                             const int* __restrict__ cols,
                             const float* __restrict__ vals,
                             const float* __restrict__ H,
                             float* __restrict__ S) {
  long long tid = (long long)blockIdx.x * blockDim.x + threadIdx.x;
  int e = (int)(tid >> 5);
  int cl = (int)(tid & 31);
  if (e >= N_EDGES) return;
  const int r = rows[e], co = cols[e];
  const float v = vals[e];
  const float* hrow = H + (long long)co * HID;
  float* srow = S + (long long)r * HID;
#pragma unroll
  for (int q = 0; q < 4; ++q) {
    int j = cl + 32 * q;
    atomicAdd(&srow[j], v * hrow[j]);
  }
}

// ---------------------------------------------------------------------------
// GEMM2: G2[100000x16] = relu(S1)[100000x128] @ W2p[128x16], fp32 WMMA.
// One wave per 16-row tile; ReLU fused into the A-operand load.
// ---------------------------------------------------------------------------
__global__ void gcn_gemm2_wmma(const float* __restrict__ S1,
                               const float* __restrict__ W2p,
                               float* __restrict__ G2) {
  const int lane = threadIdx.x & 31;
  const int wv   = threadIdx.x >> 5;
  const int mt   = blockIdx.x * 8 + wv;       // M tile index
  if (mt * 16 >= N_NODES) return;             // whole-wave exit: EXEC stays all-1
  const int half = lane >> 4;
  const int l    = lane & 15;
  const int m0   = mt * 16;

  const float* srow = S1 + (long long)(m0 + l) * HID;
  v8f c = {0.f, 0.f, 0.f, 0.f, 0.f, 0.f, 0.f, 0.f};

  for (int k0 = 0; k0 < HID; k0 += 4) {
    const int ka = k0 + 2 * half;
    v2f a;
    a.x = fmaxf(srow[ka], 0.0f);
    a.y = fmaxf(srow[ka + 1], 0.0f);
    v2f b;
    b.x = W2p[ka * NPAD + l];
    b.y = W2p[(ka + 1) * NPAD + l];
    c = __builtin_amdgcn_wmma_f32_16x16x4_f32(false, a, false, b,
                                              (short)0, c, false, false);
  }
#pragma unroll
  for (int r = 0; r < 8; ++r)
    G2[(long long)(m0 + half * 8 + r) * NPAD + l] = c[r];
}

// ---------------------------------------------------------------------------
// SpMM layer 2 (width 16, padded): 16 lanes per edge.
// ---------------------------------------------------------------------------
__global__ void gcn_spmm_cls(const int* __restrict__ rows,
                             const int* __restrict__ cols,
                             const float* __restrict__ vals,
                             const float* __restrict__ G2,
                             float* __restrict__ S2) {
  long long tid = (long long)blockIdx.x * blockDim.x + threadIdx.x;
  int e = (int)(tid >> 4);
  int j = (int)(tid & 15);
  if (e >= N_EDGES) return;
  const int r = rows[e], co = cols[e];
  const float v = vals[e];
  atomicAdd(&S2[(long long)r * NPAD + j], v * G2[(long long)co * NPAD + j]);
}

// ---------------------------------------------------------------------------
// Final: h = (relu(S2_v0) + relu(S2_v1)) / 2, log-softmax NLL, mean reduce.
// ---------------------------------------------------------------------------
__global__ void gcn_loss(const float* __restrict__ S2a, const float* __restrict__ S2b,
                         const int* __restrict__ labels, const int* __restrict__ idxt,
                         float* __restrict__ out) {
  int t = blockIdx.x * blockDim.x + threadIdx.x;
  if (t >= N_TRAIN) return;
  const int idx = idxt[t];
  const int lab = labels[idx];
  float lg[NCLS];
  float mx = -3.0e38f;
#pragma unroll
  for (int j = 0; j < NCLS; ++j) {
    float a = fmaxf(S2a[(long long)idx * NPAD + j], 0.0f);
    float b = fmaxf(S2b[(long long)idx * NPAD + j], 0.0f);
    lg[j] = 0.5f * (a + b);
    mx = fmaxf(mx, lg[j]);
  }
  float s = 0.0f;
#pragma unroll
  for (int j = 0; j < NCLS; ++j) s += expf(lg[j] - mx);
  float nll = (mx + logf(s)) - lg[lab];
  atomicAdd(out, nll * (1.0f / (float)N_TRAIN));
}

// ---------------------------------------------------------------------------
// Launch: d_in order = {x, W1, b1, W2, b2, edge_vals, edge_rows, edge_cols,
//                       labels, idx_train}.  Output = 1 float (mean NLL).
// Workspace: H1(12.8M) | S1(12.8M) | G2(1.6M) | S2 x2 (3.2M) | W2p(2048)
//            floats  ~= 122 MB.
// ---------------------------------------------------------------------------
extern "C" void kernel_launch(void* const* d_in, const int* in_sizes, int n_in,
                              void* d_out, int out_size, void* d_ws, size_t ws_size,
                              hipStream_t stream) {
  const float* x    = (const float*)d_in[0];
  const float* W1   = (const float*)d_in[1];
  const float* b1   = (const float*)d_in[2];
  const float* W2   = (const float*)d_in[3];
  const float* b2   = (const float*)d_in[4];
  const float* ev   = (const float*)d_in[5];  // [2, E]
  const int*   er   = (const int*)d_in[6];    // [2, E]
  const int*   ec   = (const int*)d_in[7];    // [2, E]
  const int*   lab  = (const int*)d_in[8];
  const int*   idxt = (const int*)d_in[9];
  float* out = (float*)d_out;

  float* H1  = (float*)d_ws;
  float* S1  = H1 + (size_t)N_NODES * HID;
  float* G2  = S1 + (size_t)N_NODES * HID;
  float* S2  = G2 + (size_t)N_NODES * NPAD;          // two views back-to-back
  float* W2p = S2 + (size_t)2 * N_NODES * NPAD;

  const int TB = 256;

  // Pad W2, zero scalar output.
  gcn_setup<<<(HID * NPAD + TB - 1) / TB, TB, 0, stream>>>(W2, W2p, out);

  // H1 = x @ W1 (shared by both adjacency views).
  gcn_gemm1_wmma<<<N_NODES / 16, TB, 0, stream>>>(x, W1, H1);

  const long long s1_total = (long long)N_NODES * HID;
  const long long s2_total = (long long)N_NODES * NPAD;

  for (int a = 0; a < 2; ++a) {
    const float* va = ev + (size_t)a * N_EDGES;
    const int*   ra = er + (size_t)a * N_EDGES;
    const int*   ca = ec + (size_t)a * N_EDGES;
    float* S2v = S2 + (size_t)a * N_NODES * NPAD;

    // S1 <- b1 broadcast; then S1 += spmm(A, H1).
    gcn_init_bias<<<(int)((s1_total + TB - 1) / TB), TB, 0, stream>>>(
        S1, b1, HID, HID, s1_total);
    gcn_spmm_hid<<<(int)(((long long)N_EDGES * 32 + TB - 1) / TB), TB, 0, stream>>>(
        ra, ca, va, H1, S1);

    // G2 = relu(S1) @ W2p.
    gcn_gemm2_wmma<<<(N_NODES / 16 + 7) / 8, TB, 0, stream>>>(S1, W2p, G2);

    // S2v <- b2 broadcast (pad cols 0); then S2v += spmm(A, G2).
    gcn_init_bias<<<(int)((s2_total + TB - 1) / TB), TB, 0, stream>>>(
        S2v, b2, NPAD, NCLS, s2_total);
    gcn_spmm_cls<<<(int)(((long long)N_EDGES * 16 + TB - 1) / TB), TB, 0, stream>>>(
        ra, ca, va, G2, S2v);
  }

  // Mean NLL over the 5000 training rows.
  gcn_loss<<<(N_TRAIN + TB - 1) / TB, TB, 0, stream>>>(S2 ,
      S2 + (size_t)N_NODES * NPAD, lab, idxt, out);
}